// weight_shift_fn_28810640622005
// MI455X (gfx1250) — compile-verified
//
#include <hip/hip_runtime.h>

// CDNA5 / gfx1250, wave32.
typedef float v2f __attribute__((ext_vector_type(2)));
typedef float v4f __attribute__((ext_vector_type(4)));
typedef float v8f __attribute__((ext_vector_type(8)));

#define NBLK 2048
#define TPB  256
#define WAVES_PER_BLK (TPB / 32)

// weight = 2^round(shift) * round(sign), exact via v_rndne_f32 + v_ldexp_f32
__device__ __forceinline__ float make_w(float sh, float sg) {
    return ldexpf(rintf(sg), (int)rintf(sh));
}

// APoT grid {0,0.1,0.2,0.3,0.4,0.6,0.8,1.0}; argmin tie -> lower index, so
// strictly-greater-than-midpoint selects the higher grid value.
__device__ __forceinline__ float quantize1(float w, float mean, float scale, float alpha) {
    float x = (w - mean) * scale;                 // (w-mean)/std/alpha
    x = fminf(fmaxf(x, -1.0f), 1.0f);
    float xa = fabsf(x);
    float q = 0.0f;
    q = (xa > 0.05f) ? 0.1f : q;
    q = (xa > 0.15f) ? 0.2f : q;
    q = (xa > 0.25f) ? 0.3f : q;
    q = (xa > 0.35f) ? 0.4f : q;
    q = (xa > 0.50f) ? 0.6f : q;
    q = (xa > 0.70f) ? 0.8f : q;
    q = (xa > 0.90f) ? 1.0f : q;
    return copysignf(q, x) * alpha;
}

// Pass 1: stream shift/sign (fills L2, RT policy), accumulate sum & sumsq.
// Reduction adds are pushed through V_WMMA_F32_16X16X4_F32 with B = ones:
// D accumulates exact f32 FMAs; total over all 256 D elements = 16 * data sum.
__global__ __launch_bounds__(TPB) void pass1_stats(
    const float* __restrict__ shift, const float* __restrict__ sgn,
    double* __restrict__ ws, int n) {
    const int tid    = blockIdx.x * blockDim.x + threadIdx.x;
    const int total  = gridDim.x * blockDim.x;
    const int n4     = n >> 2;
    const int iters  = n4 / total;               // uniform: EXEC all-1s in WMMA region

    const v4f* s4 = (const v4f*)shift;
    const v4f* g4 = (const v4f*)sgn;

    v8f csum = {};
    v8f csq  = {};
    v2f ones = {1.0f, 1.0f};

    int i = tid;
    for (int it = 0; it < iters; ++it, i += total) {
        __builtin_prefetch(&s4[i + total], 0, 3);   // global_prefetch_b8
        __builtin_prefetch(&g4[i + total], 0, 3);
        v4f sh = s4[i];
        v4f sg = g4[i];
        float w0 = make_w(sh.x, sg.x), w1 = make_w(sh.y, sg.y);
        float w2 = make_w(sh.z, sg.z), w3 = make_w(sh.w, sg.w);
        v2f a  = { w0 + w1, w2 + w3 };
        v2f a2 = { w0 * w0 + w1 * w1, w2 * w2 + w3 * w3 };
        // D = A x ones + C  (8 args: neg_a, A, neg_b, B, c_mod, C, reuse_a, reuse_b)
        csum = __builtin_amdgcn_wmma_f32_16x16x4_f32(false, a,  false, ones, (short)0, csum, false, false);
        csq  = __builtin_amdgcn_wmma_f32_16x16x4_f32(false, a2, false, ones, (short)0, csq,  false, false);
    }

    // Divergent tails (empty for 4096^2): plain double accumulation.
    double ts = 0.0, tq = 0.0;
    for (int i2 = iters * total + tid; i2 < n4; i2 += total) {
        v4f sh = s4[i2];
        v4f sg = g4[i2];
        float w0 = make_w(sh.x, sg.x), w1 = make_w(sh.y, sg.y);
        float w2 = make_w(sh.z, sg.z), w3 = make_w(sh.w, sg.w);
        ts += (double)w0 + (double)w1 + (double)w2 + (double)w3;
        tq += (double)(w0*w0) + (double)(w1*w1) + (double)(w2*w2) + (double)(w3*w3);
    }
    for (int j = n4 * 4 + tid; j < n; j += total) {
        float w = make_w(shift[j], sgn[j]);
        ts += (double)w;
        tq += (double)(w * w);
    }

    // Per-lane: fold 8 D registers; scale WMMA part by exact 1/16 (column replication).
    double ls = ts, lq = tq;
#pragma unroll
    for (int r = 0; r < 8; ++r) {
        ls += 0.0625 * (double)csum[r];
        lq += 0.0625 * (double)csq[r];
    }
    // wave32 butterfly reduction (deterministic tree)
#pragma unroll
    for (int off = 16; off >= 1; off >>= 1) {
        ls += __shfl_xor(ls, off, 32);
        lq += __shfl_xor(lq, off, 32);
    }

    __shared__ double wsum[WAVES_PER_BLK], wsq[WAVES_PER_BLK];
    const int wave = threadIdx.x >> 5;
    const int lane = threadIdx.x & 31;
    if (lane == 0) { wsum[wave] = ls; wsq[wave] = lq; }
    __syncthreads();
    if (threadIdx.x == 0) {
        double S = 0.0, Q = 0.0;
#pragma unroll
        for (int w = 0; w < WAVES_PER_BLK; ++w) { S += wsum[w]; Q += wsq[w]; }
        ws[blockIdx.x]        = S;
        ws[NBLK + blockIdx.x] = Q;
    }
}

// Pass 2: single block, fixed-order deterministic reduction of 2048 partials,
// then mean and 1/std (ddof=1) in f64.
__global__ __launch_bounds__(TPB) void pass2_finalize(double* __restrict__ ws, int n) {
    __shared__ double as[TPB], aq[TPB];
    double s = 0.0, q = 0.0;
    for (int j = threadIdx.x; j < NBLK; j += TPB) {
        s += ws[j];
        q += ws[NBLK + j];
    }
    as[threadIdx.x] = s;
    aq[threadIdx.x] = q;
    __syncthreads();
    for (int off = TPB >> 1; off >= 1; off >>= 1) {
        if ((int)threadIdx.x < off) {
            as[threadIdx.x] += as[threadIdx.x + off];
            aq[threadIdx.x] += aq[threadIdx.x + off];
        }
        __syncthreads();
    }
    if (threadIdx.x == 0) {
        double N    = (double)n;
        double S    = as[0], Q = aq[0];
        double mean = S / N;
        double var  = (Q - S * S / N) / (N - 1.0);
        ws[2 * NBLK]     = mean;
        ws[2 * NBLK + 1] = 1.0 / sqrt(var);
    }
}

// Pass 3: re-stream inputs (L2-resident from pass 1), quantize, NT stores so
// the 67MB of output doesn't evict the inputs from L2.
__global__ __launch_bounds__(TPB) void pass3_quant(
    const float* __restrict__ shift, const float* __restrict__ sgn,
    const float* __restrict__ alpha_p, const double* __restrict__ ws,
    float* __restrict__ out, int n) {
    const int tid   = blockIdx.x * blockDim.x + threadIdx.x;
    const int total = gridDim.x * blockDim.x;
    const int n4    = n >> 2;

    const float alpha = alpha_p[0];
    const float mean  = (float)ws[2 * NBLK];
    const float scale = (float)(ws[2 * NBLK + 1] / (double)alpha);  // 1/(std*alpha)

    const v4f* s4 = (const v4f*)shift;
    const v4f* g4 = (const v4f*)sgn;
    v4f*       o4 = (v4f*)out;

    for (int i = tid; i < n4; i += total) {
        __builtin_prefetch(&s4[i + total], 0, 3);
        __builtin_prefetch(&g4[i + total], 0, 3);
        v4f sh = s4[i];
        v4f sg = g4[i];
        v4f r;
        r.x = quantize1(make_w(sh.x, sg.x), mean, scale, alpha);
        r.y = quantize1(make_w(sh.y, sg.y), mean, scale, alpha);
        r.z = quantize1(make_w(sh.z, sg.z), mean, scale, alpha);
        r.w = quantize1(make_w(sh.w, sg.w), mean, scale, alpha);
        __builtin_nontemporal_store(r, &o4[i]);     // TH=NT 128-bit store
    }
    for (int j = n4 * 4 + tid; j < n; j += total) {
        float r = quantize1(make_w(shift[j], sgn[j]), mean, scale, alpha);
        __builtin_nontemporal_store(r, &out[j]);
    }
}

extern "C" void kernel_launch(void* const* d_in, const int* in_sizes, int n_in,
                              void* d_out, int out_size, void* d_ws, size_t ws_size,
                              hipStream_t stream) {
    const float* shift = (const float*)d_in[0];
    const float* sgn   = (const float*)d_in[1];
    const float* alpha = (const float*)d_in[2];
    float*       out   = (float*)d_out;
    double*      ws    = (double*)d_ws;   // needs 2*NBLK+2 doubles (~32.8 KB)
    const int    n     = in_sizes[0];

    pass1_stats<<<NBLK, TPB, 0, stream>>>(shift, sgn, ws, n);
    pass2_finalize<<<1, TPB, 0, stream>>>(ws, n);
    pass3_quant<<<NBLK, TPB, 0, stream>>>(shift, sgn, alpha, ws, out, n);
}